// WGCN_29068338659496
// MI455X (gfx1250) — compile-verified
//
#include <hip/hip_runtime.h>

// ---------------------------------------------------------------------------
// WGCN pipeline for MI455X (gfx1250, wave32).
// Dense GEMMs use V_WMMA_F32_16X16X4_F32 (exact fp32 matrix pipe), with
// branchless clamped loads and two interleaved accumulator chains to avoid
// WMMA->WMMA RAW hazards. Conv1d is algebraically folded over the tiny token
// vocab (5 / 65), cutting 48 GFLOP -> ~0.8 GFLOP. Edge aggregation is
// atomics-based (memory bound: ~3 GB of traffic at 23.3 TB/s).
// ---------------------------------------------------------------------------

#define NNODES 50000
#define NEDGES 1600000
#define NGRAPH 128
#define NBATCH 128
#define MLEN   3000
#define EDIM   128

typedef float v2f __attribute__((ext_vector_type(2)));
typedef float v8f __attribute__((ext_vector_type(8)));

// ---------------------------------------------------------------------------
// Generic fp32 GEMM: C[M,N] = A[M,K] @ B[K,N] (+bias) (+relu), row-major.
// One 16x16 C tile per wave via V_WMMA_F32_16X16X4_F32.
// Fragment layouts per CDNA5 ISA 7.12.2:
//   A 16x4 : lane(half=l>>4, m=l&15) holds A[m, k0+2*half+j] in vj
//   B 4x16 : lane holds B[k0+2*half+j, n=l&15] in vj
//   C 16x16: vr holds C[r+8*half, n=l&15]
// Out-of-range lanes load from clamped (valid) addresses and are zeroed by
// multiplying with ascale/bscale -- no exec-mask branches in the hot loop.
// ---------------------------------------------------------------------------
__global__ void __launch_bounds__(128)
gemm_f32_wmma(const float* __restrict__ A, const float* __restrict__ B,
              const float* __restrict__ bias, float* __restrict__ C,
              int M, int N, int K, int relu)
{
    const int tilesN = (N + 15) >> 4;
    const int tilesM = (M + 15) >> 4;
    const int wid = blockIdx.x * 4 + (threadIdx.x >> 5);
    if (wid >= tilesM * tilesN) return;
    const int tm = wid / tilesN;
    const int tn = wid % tilesN;
    const int lane = threadIdx.x & 31;
    const int half = lane >> 4;
    const int r16  = lane & 15;
    const int hof  = 2 * half;
    const int m = tm * 16 + r16;   // A row owned by this lane
    const int n = tn * 16 + r16;   // B col owned by this lane

    const float ascale = (m < M) ? 1.0f : 0.0f;
    const float bscale = (n < N) ? 1.0f : 0.0f;
    const float* Ap = A + (size_t)(m < M ? m : M - 1) * K;   // clamped row
    const float* Bp = B + (n < N ? n : N - 1);               // clamped col

    v8f acc0 = {};
    v8f acc1 = {};
    int k0 = 0;

    // main loop: two independent WMMA chains, no bounds checks needed
    for (; k0 + 8 <= K; k0 += 8) {
        __builtin_prefetch(Ap + k0 + 64);                    // speculative, dropped if OOB
        __builtin_prefetch(Bp + (size_t)(k0 + 16) * N);
        {
            const int ka = k0 + hof;
            v2f a, b;
            a[0] = Ap[ka]     * ascale;
            a[1] = Ap[ka + 1] * ascale;
            b[0] = Bp[(size_t)ka * N]       * bscale;
            b[1] = Bp[(size_t)(ka + 1) * N] * bscale;
            acc0 = __builtin_amdgcn_wmma_f32_16x16x4_f32(
                false, a, false, b, (short)0, acc0, false, false);
        }
        {
            const int kb = k0 + 4 + hof;
            v2f a, b;
            a[0] = Ap[kb]     * ascale;
            a[1] = Ap[kb + 1] * ascale;
            b[0] = Bp[(size_t)kb * N]       * bscale;
            b[1] = Bp[(size_t)(kb + 1) * N] * bscale;
            acc1 = __builtin_amdgcn_wmma_f32_16x16x4_f32(
                false, a, false, b, (short)0, acc1, false, false);
        }
    }
    for (; k0 + 4 <= K; k0 += 4) {
        const int ka = k0 + hof;
        v2f a, b;
        a[0] = Ap[ka]     * ascale;
        a[1] = Ap[ka + 1] * ascale;
        b[0] = Bp[(size_t)ka * N]       * bscale;
        b[1] = Bp[(size_t)(ka + 1) * N] * bscale;
        acc0 = __builtin_amdgcn_wmma_f32_16x16x4_f32(
            false, a, false, b, (short)0, acc0, false, false);
    }
    if (k0 < K) {   // K % 4 tail: clamp addresses, zero via select (branchless)
        const int ka = k0 + hof;
        const int kc0 = (ka     < K) ? ka     : K - 1;
        const int kc1 = (ka + 1 < K) ? ka + 1 : K - 1;
        const float a0 = Ap[kc0], a1 = Ap[kc1];
        const float b0 = Bp[(size_t)kc0 * N], b1 = Bp[(size_t)kc1 * N];
        v2f a, b;
        a[0] = (ka     < K) ? a0 * ascale : 0.0f;
        a[1] = (ka + 1 < K) ? a1 * ascale : 0.0f;
        b[0] = (ka     < K) ? b0 * bscale : 0.0f;
        b[1] = (ka + 1 < K) ? b1 * bscale : 0.0f;
        acc1 = __builtin_amdgcn_wmma_f32_16x16x4_f32(
            false, a, false, b, (short)0, acc1, false, false);
    }

    const v8f acc = acc0 + acc1;
    for (int r = 0; r < 8; ++r) {
        const int mo = tm * 16 + r + 8 * half;
        const int no = tn * 16 + r16;
        if (mo < M && no < N) {
            float v = acc[r];
            if (bias) v += bias[no];
            if (relu) v = fmaxf(v, 0.0f);
            C[(size_t)mo * N + no] = v;
        }
    }
}

static void launch_gemm(const float* A, const float* B, const float* bias,
                        float* C, int M, int N, int K, int relu,
                        hipStream_t stream)
{
    const int tiles = ((M + 15) / 16) * ((N + 15) / 16);
    const int blocks = (tiles + 3) / 4;
    gemm_f32_wmma<<<blocks, 128, 0, stream>>>(A, B, bias, C, M, N, K, relu);
}

// ---------------------------------------------------------------------------
// GCN helpers
// ---------------------------------------------------------------------------
__global__ void deg_accum(const int* __restrict__ ei, const float* __restrict__ ew,
                          float* __restrict__ deg1, float* __restrict__ deg23, int E)
{
    const int e = blockIdx.x * blockDim.x + threadIdx.x;
    if (e >= E) return;
    const int d = ei[E + e];
    atomicAdd(&deg1[d], ew[e]);
    atomicAdd(&deg23[d], 1.0f);
}

__global__ void deg_to_dis(float* __restrict__ d1, float* __restrict__ d23, int n)
{
    const int i = blockIdx.x * blockDim.x + threadIdx.x;
    if (i >= n) return;
    d1[i]  = rsqrtf(d1[i]  + 1.0f);   // self-loop weight 1
    d23[i] = rsqrtf(d23[i] + 1.0f);
}

// agg[n,f] = xw[n,f]*dis[n]^2 + bias[f]   (self-loop term + layer bias)
__global__ void self_bias(const float* __restrict__ xw, const float* __restrict__ dis,
                          const float* __restrict__ b, float* __restrict__ agg,
                          int N, int F)
{
    const int idx = blockIdx.x * blockDim.x + threadIdx.x;
    if (idx >= N * F) return;
    const int n = idx / F, f = idx % F;
    const float d = dis[n];
    agg[idx] = xw[idx] * d * d + b[f];
}

// agg[dst,f] += xw[src,f] * dis[src]*w*dis[dst]
__global__ void edge_scatter(const int* __restrict__ ei, const float* __restrict__ ew,
                             const float* __restrict__ dis, const float* __restrict__ xw,
                             float* __restrict__ agg, int E, int F)
{
    const int e = blockIdx.x * blockDim.x + threadIdx.x;
    if (e >= E) return;
    const int f = blockIdx.y;
    const int s = ei[e];
    const int d = ei[E + e];
    const float w = ew ? ew[e] : 1.0f;
    const float norm = dis[s] * w * dis[d];
    atomicAdd(&agg[(size_t)d * F + f], xw[(size_t)s * F + f] * norm);
}

// per-column sum / sumsq into stats[0:F], stats[F:2F]
__global__ void bn_stats(const float* __restrict__ x, float* __restrict__ stats,
                         int N, int F)
{
    __shared__ float s1[256], s2[256];
    const int f = blockIdx.y;
    const int n = blockIdx.x * 256 + threadIdx.x;
    const float v = (n < N) ? x[(size_t)n * F + f] : 0.0f;
    s1[threadIdx.x] = v;
    s2[threadIdx.x] = v * v;
    __syncthreads();
    for (int s = 128; s > 0; s >>= 1) {
        if (threadIdx.x < s) {
            s1[threadIdx.x] += s1[threadIdx.x + s];
            s2[threadIdx.x] += s2[threadIdx.x + s];
        }
        __syncthreads();
    }
    if (threadIdx.x == 0) {
        atomicAdd(&stats[f],     s1[0]);
        atomicAdd(&stats[F + f], s2[0]);
    }
}

__global__ void bn_apply_relu(const float* __restrict__ x, const float* __restrict__ stats,
                              const float* __restrict__ g, const float* __restrict__ b,
                              float* __restrict__ y, int N, int F)
{
    const int idx = blockIdx.x * blockDim.x + threadIdx.x;
    if (idx >= N * F) return;
    const int f = idx % F;
    const float invN = 1.0f / (float)N;
    const float mean = stats[f] * invN;
    const float var  = stats[F + f] * invN - mean * mean;
    const float r = (x[idx] - mean) * rsqrtf(var + 1e-5f) * g[f] + b[f];
    y[idx] = fmaxf(r, 0.0f);
}

__global__ void pool_accum(const float* __restrict__ x, const int* __restrict__ batch,
                           float* __restrict__ pooled, float* __restrict__ cnt,
                           int N, int F)
{
    const int idx = blockIdx.x * blockDim.x + threadIdx.x;
    if (idx >= N * F) return;
    const int n = idx / F, f = idx % F;
    const int g = batch[n];
    atomicAdd(&pooled[(size_t)g * F + f], x[idx]);
    if (f == 0) atomicAdd(&cnt[g], 1.0f);
}

__global__ void pool_div(float* __restrict__ pooled, const float* __restrict__ cnt,
                         int total, int F)
{
    const int idx = blockIdx.x * blockDim.x + threadIdx.x;
    if (idx >= total) return;
    pooled[idx] /= fmaxf(cnt[idx / F], 1.0f);
}

// ---------------------------------------------------------------------------
// RNA conv branch, folded over vocab:
//   y[b,o,l] = cb[o] + sum_t sum_k Agg[t,o,k] * emb[t, l+k]
//   Agg[t,o,k] = sum_{c : tok[b,c]==t} W[o,c,k]
// One workgroup per batch element; thread tid owns (o=tid>>3, k=tid&7) so the
// fold needs no atomics. Agg lives in dynamic LDS (V*256 floats, <=65 KB --
// fine within gfx1250's 320 KB WGP LDS).
// ---------------------------------------------------------------------------
__global__ void __launch_bounds__(256)
rna_conv_fold(const int* __restrict__ tok, const float* __restrict__ emb,
              const float* __restrict__ W, const float* __restrict__ cb,
              float* __restrict__ y, int C, int V)
{
    extern __shared__ float Ag[];                     // [V][32][8]
    const int b = blockIdx.x;
    const int tid = threadIdx.x;
    for (int i = tid; i < V * 256; i += 256) Ag[i] = 0.0f;
    __syncthreads();

    const int o = tid >> 3, k = tid & 7;
    const float* Wok = W + (size_t)o * C * 8 + k;     // W[o,c,k]
    const int* tb = tok + (size_t)b * C;
    for (int c = 0; c < C; ++c) {
        const int t = tb[c];                          // uniform per iteration
        Ag[t * 256 + tid] += Wok[(size_t)c * 8];
    }
    __syncthreads();

    for (int oi = tid; oi < 32 * 121; oi += 256) {
        const int oo = oi / 121, l = oi % 121;
        float acc = cb[oo];
        for (int t = 0; t < V; ++t) {
            const float* a = &Ag[t * 256 + oo * 8];
            const float* e = &emb[t * EDIM + l];
            #pragma unroll
            for (int kk = 0; kk < 8; ++kk) acc += a[kk] * e[kk];
        }
        y[(size_t)b * 3872 + oo * 121 + l] = acc;     // matches reshape(B,-1)
    }
}

__global__ void combine_rna(const float* __restrict__ xr_fc, float* __restrict__ out)
{
    const int idx = blockIdx.x * blockDim.x + threadIdx.x;
    if (idx >= NBATCH * 128) return;
    out[idx] = 0.5f * (xr_fc[idx] + xr_fc[NBATCH * 128 + idx]);
}

// ---------------------------------------------------------------------------
// Host side
// ---------------------------------------------------------------------------
static void gcn_layer(const float* x_in, int Fin, int Fout,
                      const float* Wl, const float* bl,
                      const float* dis, const float* ew,
                      const float* bn_g, const float* bn_b,
                      const int* ei, float* xw, float* agg, float* xp,
                      float* stats, hipStream_t stream)
{
    launch_gemm(x_in, Wl, nullptr, xw, NNODES, Fout, Fin, 0, stream);
    {
        const int tot = NNODES * Fout;
        self_bias<<<(tot + 255) / 256, 256, 0, stream>>>(xw, dis, bl, agg, NNODES, Fout);
    }
    {
        dim3 grid((NEDGES + 255) / 256, Fout);
        edge_scatter<<<grid, 256, 0, stream>>>(ei, ew, dis, xw, agg, NEDGES, Fout);
    }
    hipMemsetAsync(stats, 0, 2 * 132 * sizeof(float), stream);
    {
        dim3 grid((NNODES + 255) / 256, Fout);
        bn_stats<<<grid, 256, 0, stream>>>(agg, stats, NNODES, Fout);
    }
    {
        const int tot = NNODES * Fout;
        bn_apply_relu<<<(tot + 255) / 256, 256, 0, stream>>>(agg, stats, bn_g, bn_b,
                                                             xp, NNODES, Fout);
    }
}

extern "C" void kernel_launch(void* const* d_in, const int* in_sizes, int n_in,
                              void* d_out, int out_size, void* d_ws, size_t ws_size,
                              hipStream_t stream)
{
    (void)in_sizes; (void)n_in; (void)out_size; (void)ws_size;
    // inputs in setup_inputs() order
    const float* pro_x      = (const float*)d_in[0];
    const int*   edge_index = (const int*)  d_in[1];
    const float* edge_w     = (const float*)d_in[2];
    const int*   pro_batch  = (const int*)  d_in[3];
    const int*   rna_global = (const int*)  d_in[4];
    const int*   rna_local  = (const int*)  d_in[5];
    const float* emb1    = (const float*)d_in[6];
    const float* emb2    = (const float*)d_in[7];
    const float* convW1  = (const float*)d_in[8];
    const float* convb1  = (const float*)d_in[9];
    const float* convW2  = (const float*)d_in[10];
    const float* convb2  = (const float*)d_in[11];
    const float* fcxr_W  = (const float*)d_in[12];
    const float* fcxr_b  = (const float*)d_in[13];
    const float* gcn1_W  = (const float*)d_in[14];
    const float* gcn1_b  = (const float*)d_in[15];
    const float* gcn2_W  = (const float*)d_in[16];
    const float* gcn2_b  = (const float*)d_in[17];
    const float* gcn3_W  = (const float*)d_in[18];
    const float* gcn3_b  = (const float*)d_in[19];
    const float* bn1_g   = (const float*)d_in[20];
    const float* bn1_b   = (const float*)d_in[21];
    const float* bn2_g   = (const float*)d_in[22];
    const float* bn2_b   = (const float*)d_in[23];
    const float* bn3_g   = (const float*)d_in[24];
    const float* bn3_b   = (const float*)d_in[25];
    const float* fcg1_W  = (const float*)d_in[26];
    const float* fcg1_b  = (const float*)d_in[27];
    const float* fcg2_W  = (const float*)d_in[28];
    const float* fcg2_b  = (const float*)d_in[29];

    float* out = (float*)d_out;            // [0:16384) xc_rna, [16384:32768) xp_out

    // workspace carve-up (floats)
    float* W = (float*)d_ws;
    size_t o = 0;
    float* xw     = W + o; o += (size_t)NNODES * 132;
    float* agg    = W + o; o += (size_t)NNODES * 132;
    float* xp     = W + o; o += (size_t)NNODES * 132;
    float* dis1   = W + o; o += NNODES;
    float* dis23  = W + o; o += NNODES;
    float* stats  = W + o; o += 2 * 132;
    float* pooled = W + o; o += (size_t)NGRAPH * 132;
    float* cnt    = W + o; o += NGRAPH;
    float* xr_y   = W + o; o += (size_t)2 * NBATCH * 3872;   // stacked [256,3872]
    float* xr_fc  = W + o; o += (size_t)2 * NBATCH * 128;
    float* hbuf   = W + o; o += (size_t)NBATCH * 1024;

    // ---------------- RNA branch ----------------
    rna_conv_fold<<<NBATCH, 256, 5  * 256 * sizeof(float), stream>>>(
        rna_global, emb1, convW1, convb1, xr_y, MLEN, 5);
    rna_conv_fold<<<NBATCH, 256, 65 * 256 * sizeof(float), stream>>>(
        rna_local, emb2, convW2, convb2, xr_y + (size_t)NBATCH * 3872, MLEN - 2, 65);
    // shared fc_xr on stacked [256, 3872]
    launch_gemm(xr_y, fcxr_W, fcxr_b, xr_fc, 2 * NBATCH, 128, 3872, 0, stream);
    combine_rna<<<(NBATCH * 128 + 255) / 256, 256, 0, stream>>>(xr_fc, out);

    // ---------------- GCN degrees ----------------
    hipMemsetAsync(dis1,  0, NNODES * sizeof(float), stream);
    hipMemsetAsync(dis23, 0, NNODES * sizeof(float), stream);
    deg_accum<<<(NEDGES + 255) / 256, 256, 0, stream>>>(edge_index, edge_w, dis1, dis23, NEDGES);
    deg_to_dis<<<(NNODES + 255) / 256, 256, 0, stream>>>(dis1, dis23, NNODES);

    // ---------------- GCN layers ----------------
    gcn_layer(pro_x, 33, 33,  gcn1_W, gcn1_b, dis1,  edge_w,  bn1_g, bn1_b,
              edge_index, xw, agg, xp, stats, stream);
    gcn_layer(xp,    33, 66,  gcn2_W, gcn2_b, dis23, nullptr, bn2_g, bn2_b,
              edge_index, xw, agg, xp, stats, stream);
    gcn_layer(xp,    66, 132, gcn3_W, gcn3_b, dis23, nullptr, bn3_g, bn3_b,
              edge_index, xw, agg, xp, stats, stream);

    // ---------------- pooling + head ----------------
    hipMemsetAsync(pooled, 0, (size_t)NGRAPH * 132 * sizeof(float), stream);
    hipMemsetAsync(cnt,    0, NGRAPH * sizeof(float), stream);
    {
        const int tot = NNODES * 132;
        pool_accum<<<(tot + 255) / 256, 256, 0, stream>>>(xp, pro_batch, pooled, cnt,
                                                          NNODES, 132);
    }
    {
        const int tot = NGRAPH * 132;
        pool_div<<<(tot + 255) / 256, 256, 0, stream>>>(pooled, cnt, tot, 132);
    }
    launch_gemm(pooled, fcg1_W, fcg1_b, hbuf, NGRAPH, 1024, 132, 1, stream);
    launch_gemm(hbuf, fcg2_W, fcg2_b, out + NBATCH * 128, NGRAPH, 128, 1024, 0, stream);
}